// QueryAndGroup_70385924047160
// MI455X (gfx1250) — compile-verified
//
#include <hip/hip_runtime.h>
#include <stdint.h>

typedef __attribute__((ext_vector_type(2))) float v2f;
typedef __attribute__((ext_vector_type(8))) float v8f;

#define RADIUS2 0.01f          // 0.1^2
#define NSAMPLE 32
#define WAVES_PER_BLOCK 2

__device__ __forceinline__ float lane_bcast(float v, int srcLane) {
    return __int_as_float(__builtin_amdgcn_ds_bpermute(srcLane << 2, __float_as_int(v)));
}

__global__ __launch_bounds__(WAVES_PER_BLOCK * 32)
void ballquery_group_wmma(const float* __restrict__ xyz,
                          const float* __restrict__ new_xyz,
                          const float* __restrict__ features,
                          float* __restrict__ out,
                          int B, int N, int P, int C) {
    __shared__ int s_idx[WAVES_PER_BLOCK][16][NSAMPLE + 1];

    const int wave = threadIdx.x >> 5;
    const int lane = threadIdx.x & 31;
    const int tilesPerBatch = P >> 4;                       // 16 queries per wave
    const int tile = blockIdx.x * WAVES_PER_BLOCK + wave;
    if (tile >= B * tilesPerBatch) return;
    const int b = tile / tilesPerBatch;
    const int qbase = (tile - b * tilesPerBatch) << 4;

    const float* xyzb = xyz     + (size_t)b * N * 3;
    const float* newb = new_xyz + (size_t)b * P * 3;

    // ---- A matrix: 16 query points. 16x4 f32 layout:
    //   VGPR0: lanes0-15 K=0, lanes16-31 K=2 ; VGPR1: lanes0-15 K=1, lanes16-31 K=3
    const int  row    = lane & 15;
    const bool hiHalf = (lane & 16) != 0;
    const float* qp = newb + (size_t)(qbase + row) * 3;
    const float qx = qp[0], qy = qp[1], qz = qp[2];
    const float q2 = qx * qx + qy * qy + qz * qz;

    v2f amat;
    amat.x = hiHalf ? qz   : qx;
    amat.y = hiHalf ? 0.0f : qy;

    // Row bias folded into the accumulator: 0.5*(r^2 - |q_row|^2) for the row
    // this C-register r holds in this lane half (row = r, or r+8 in hi half).
    float q2h[8];
#pragma unroll
    for (int r = 0; r < 8; ++r) {
        const int src = r + (hiHalf ? 8 : 0);
        q2h[r] = 0.5f * (RADIUS2 - lane_bcast(q2, src));
    }

    int cnt = 0;
    int first = 0;

    // Process one 16-column tile held in registers: WMMA + ballots + hit append.
    auto processTile = [&](float xx, float xy, float xz, int base) {
        const float xn2h = 0.5f * (xx * xx + xy * xy + xz * xz);
        v2f bmat;
        bmat.x = hiHalf ? xz   : xx;
        bmat.y = hiHalf ? 0.0f : xy;

        // C init: 0.5*(r^2 - |q|^2 - |x|^2) -> D = C + q.x > 0  <=>  d2 < r^2
        v8f c;
#pragma unroll
        for (int r = 0; r < 8; ++r) c[r] = q2h[r] - xn2h;

        c = __builtin_amdgcn_wmma_f32_16x16x4_f32(false, amat, false, bmat,
                                                  (short)0, c, false, false);

        const uint32_t m0 = __builtin_amdgcn_ballot_w32(c[0] > 0.0f);
        const uint32_t m1 = __builtin_amdgcn_ballot_w32(c[1] > 0.0f);
        const uint32_t m2 = __builtin_amdgcn_ballot_w32(c[2] > 0.0f);
        const uint32_t m3 = __builtin_amdgcn_ballot_w32(c[3] > 0.0f);
        const uint32_t m4 = __builtin_amdgcn_ballot_w32(c[4] > 0.0f);
        const uint32_t m5 = __builtin_amdgcn_ballot_w32(c[5] > 0.0f);
        const uint32_t m6 = __builtin_amdgcn_ballot_w32(c[6] > 0.0f);
        const uint32_t m7 = __builtin_amdgcn_ballot_w32(c[7] > 0.0f);

        if (lane < 16) {
            // lane l owns query row l; its hits live in ballot (l&7),
            // low 16 bits for rows 0-7, high 16 bits for rows 8-15.
            const int r = lane & 7;
            const uint32_t s01 = (r & 1) ? m1 : m0;
            const uint32_t s23 = (r & 1) ? m3 : m2;
            const uint32_t s45 = (r & 1) ? m5 : m4;
            const uint32_t s67 = (r & 1) ? m7 : m6;
            const uint32_t s03 = (r & 2) ? s23 : s01;
            const uint32_t s47 = (r & 2) ? s67 : s45;
            const uint32_t mm  = (r & 4) ? s47 : s03;
            uint32_t bits = (lane < 8) ? (mm & 0xFFFFu) : (mm >> 16);
            while (bits && cnt < NSAMPLE) {
                const int j = __builtin_ctz(bits);
                bits &= bits - 1;
                const int nidx = base + j;
                if (cnt == 0) first = nidx;
                s_idx[wave][lane][cnt] = nidx;
                ++cnt;
            }
        }
    };

    // ---- software-pipelined scan: 32 columns (two 16-wide tiles) per step,
    // next pair prefetched one step ahead so the s_wait lands after ~2 tiles
    // of WMMA/ballot work instead of immediately after the load.
    const float* basep = xyzb + (size_t)row * 3;   // safe dummy target
    {
        const float* p0 = basep;                   // tile 0
        const float* p1 = basep + 48;              // tile 16
        float c0x = p0[0], c0y = p0[1], c0z = p0[2];
        float c1x = p1[0], c1y = p1[1], c1z = p1[2];

        for (int n0 = 0; n0 < N; n0 += 32) {
            const float* q0 = (n0 + 32 < N) ? (basep + (size_t)(n0 + 32) * 3) : basep;
            const float* q1 = (n0 + 48 < N) ? (basep + (size_t)(n0 + 48) * 3) : basep;
            const float nx0 = q0[0], ny0 = q0[1], nz0 = q0[2];   // prefetch
            const float nx1 = q1[0], ny1 = q1[1], nz1 = q1[2];   // prefetch

            processTile(c0x, c0y, c0z, n0);
            processTile(c1x, c1y, c1z, n0 + 16);

            const bool done = (lane >= 16) || (cnt >= NSAMPLE);
            if (__builtin_amdgcn_ballot_w32(done) == 0xFFFFFFFFu) break;

            c0x = nx0; c0y = ny0; c0z = nz0;
            c1x = nx1; c1y = ny1; c1z = nz1;
        }
    }

    // pad remaining slots with the first in-ball index (0 if none found)
    if (lane < 16) {
        for (int c2 = cnt; c2 < NSAMPLE; ++c2) s_idx[wave][lane][c2] = first;
    }

    // ---- grouping: lane = sample slot, coalesced 128B non-temporal stores
    const float* featb    = features + (size_t)b * C * N;
    const size_t outB     = (size_t)b * (3 + C) * P * NSAMPLE;
    const size_t chStride = (size_t)P * NSAMPLE;

#pragma unroll 1
    for (int q = 0; q < 16; ++q) {
        const int gq = qbase + q;
        const int id = s_idx[wave][q][lane];
        const float* pp = xyzb + (size_t)id * 3;
        const float px = pp[0], py = pp[1], pz = pp[2];
        const float* cp = newb + (size_t)gq * 3;
        const float cx = cp[0], cy = cp[1], cz = cp[2];
        float* o0 = out + outB + (size_t)gq * NSAMPLE + lane;
        __builtin_nontemporal_store(px - cx, o0);
        __builtin_nontemporal_store(py - cy, o0 + chStride);
        __builtin_nontemporal_store(pz - cz, o0 + 2 * chStride);
        const float* fptr = featb + id;
        float* oc = o0 + 3 * chStride;
#pragma unroll 16
        for (int ch = 0; ch < C; ++ch) {
            __builtin_nontemporal_store(fptr[(size_t)ch * N], oc + (size_t)ch * chStride);
        }
    }
}

extern "C" void kernel_launch(void* const* d_in, const int* in_sizes, int n_in,
                              void* d_out, int out_size, void* d_ws, size_t ws_size,
                              hipStream_t stream) {
    (void)n_in; (void)d_ws; (void)ws_size; (void)out_size;
    const float* xyz      = (const float*)d_in[0];
    const float* new_xyz  = (const float*)d_in[1];
    const float* features = (const float*)d_in[2];
    float* out = (float*)d_out;

    const int B = 2;
    const int N = in_sizes[0] / (3 * B);   // 16384
    const int P = in_sizes[1] / (3 * B);   // 4096
    const int C = in_sizes[2] / (B * N);   // 64

    const int tiles  = B * (P >> 4);
    const int blocks = (tiles + WAVES_PER_BLOCK - 1) / WAVES_PER_BLOCK;
    ballquery_group_wmma<<<blocks, WAVES_PER_BLOCK * 32, 0, stream>>>(
        xyz, new_xyz, features, out, B, N, P, C);
}